// R_Critic_63273458205196
// MI455X (gfx1250) — compile-verified
//
#include <hip/hip_runtime.h>

typedef __attribute__((ext_vector_type(2))) float v2f;
typedef __attribute__((ext_vector_type(8))) float v8f;

#define R_      16
#define NUMROWS 33
#define ROWLEN  11
#define F_      11
#define H_      256
#define E_      2048
#define B_      (E_*R_)             // 32768
#define IN_     (R_*F_)             // 176
#define D_      (R_*NUMROWS*ROWLEN) // 5808
#define G_      2                   // envs per workgroup (M=32 rows)

static __device__ __forceinline__ v8f wmma_f32(v2f a, v2f b, v8f c) {
    return __builtin_amdgcn_wmma_f32_16x16x4_f32(false, a, false, b, (short)0, c, false, false);
}

__global__ __launch_bounds__(256) void critic_fused(
    const float* __restrict__ cent_obs,
    const float* __restrict__ ln_in_g, const float* __restrict__ ln_in_b,
    const float* __restrict__ W1, const float* __restrict__ bias1,
    const float* __restrict__ ln1_g, const float* __restrict__ ln1_b,
    const float* __restrict__ W2, const float* __restrict__ bias2,
    const float* __restrict__ ln2_g, const float* __restrict__ ln2_b,
    const float* __restrict__ Wv, const float* __restrict__ bv,
    float* __restrict__ values)
{
    __shared__ float env_s[G_*R_*F_];     // 2 envs x 16 robots x 11 features
    __shared__ int   ord_s[G_*R_*R_];     // argsort order per robot
    __shared__ float xb [G_*R_*IN_];      // 2 x 16 x 176 gathered + LN'd input
    __shared__ float h1s[G_*R_*H_];       // 2 x 16 x 256
    __shared__ float h2s[G_*R_*H_];       // 2 x 16 x 256
    __shared__ float mu_s[G_*R_], rs_s[G_*R_];
    __shared__ float red[256];

    const int tid  = threadIdx.x;
    const int e0   = blockIdx.x * G_;     // first env of this workgroup
    const int lane = tid & 31;
    const int wave = tid >> 5;

    // ---- Step 1: load both envs' robots (2 x 16 x 11 = 352 floats) ----
    for (int idx = tid; idx < G_*R_*F_; idx += 256) {
        int g = idx / (R_*F_), rem = idx % (R_*F_);
        int r = rem / F_, f = rem % F_;
        env_s[idx] = cent_obs[(size_t)(e0+g) * R_ * D_ + (size_t)r * (NUMROWS*ROWLEN) + f];
    }
    __syncthreads();

    // ---- Step 2: per-robot stable argsort by distance (32 lanes: 2 envs x 16) ----
    if (tid < G_*R_) {
        const int g = tid >> 4, i = tid & 15;
        const float* env = env_s + g*R_*F_;
        const float px = env[i*F_ + (F_-2)];
        const float py = env[i*F_ + (F_-1)];
        float d[R_];
        for (int j = 0; j < R_; ++j) {
            float dx = env[j*F_ + (F_-2)] - px;
            float dy = env[j*F_ + (F_-1)] - py;
            d[j] = sqrtf(dx*dx + dy*dy + 1e-8f);
        }
        unsigned used = 0u;
        for (int k = 0; k < R_; ++k) {
            int best = -1; float bd = 0.f;
            for (int j = 0; j < R_; ++j) {
                if (used & (1u << j)) continue;
                if (best < 0 || d[j] < bd) { best = j; bd = d[j]; }
            }
            used |= 1u << best;
            ord_s[g*R_*R_ + i*R_ + k] = best;
        }
    }
    __syncthreads();

    // ---- Step 3: gather neighbors into xb (2 x 16 x 176) ----
    {
        int i = tid >> 4, k = tid & 15;
        #pragma unroll
        for (int g = 0; g < G_; ++g) {
            int src = ord_s[g*R_*R_ + i*R_ + k] * F_;
            #pragma unroll
            for (int f = 0; f < F_; ++f)
                xb[g*R_*IN_ + i*IN_ + k*F_ + f] = env_s[g*R_*F_ + src + f];
        }
    }
    __syncthreads();

    // ---- Step 4: LayerNorm over IN_ ----
    if (tid < G_*R_) {
        int g = tid >> 4, r = tid & 15;
        const float* row = xb + g*R_*IN_ + r*IN_;
        float s = 0.f, ss = 0.f;
        for (int c = 0; c < IN_; ++c) { float v = row[c]; s += v; ss += v*v; }
        float mu = s * (1.0f/IN_);
        float var = ss * (1.0f/IN_) - mu*mu;
        mu_s[tid] = mu; rs_s[tid] = rsqrtf(var + 1e-5f);
    }
    __syncthreads();
    #pragma unroll
    for (int j = 0; j < 22; ++j) {           // 256*22 == 5632 == 2*16*176
        int idx = tid * 22 + j;
        int g = idx / (R_*IN_), rem = idx % (R_*IN_);
        int r = rem / IN_, c = rem % IN_;
        xb[idx] = (xb[idx] - mu_s[g*R_+r]) * rs_s[g*R_+r] * ln_in_g[c] + ln_in_b[c];
    }
    __syncthreads();

    const int m   = lane & 15;          // A-matrix row for this lane
    const int kh  = (lane >> 4) * 2;    // K sub-offset: lanes 0-15 -> {0,1}, 16-31 -> {2,3}
    const int mh  = (lane >> 4) * 8;    // D-matrix M offset
    const int nl  = lane & 15;          // D/B-matrix column within tile
    const int n0  = wave*16 + nl;       // this wave's two N-tiles
    const int n1  = (wave+8)*16 + nl;

    // ---- Step 5: GEMM1 (2x 16x176)@(176x256) fp32 WMMA, fused bias+ReLU ----
    {
        v8f acc00 = {}, acc01 = {}, acc10 = {}, acc11 = {};
        for (int k = 0; k < IN_; k += 4) {
            v2f a0; a0.x = xb[0*R_*IN_ + m*IN_ + k + kh]; a0.y = xb[0*R_*IN_ + m*IN_ + k + kh + 1];
            v2f a1; a1.x = xb[1*R_*IN_ + m*IN_ + k + kh]; a1.y = xb[1*R_*IN_ + m*IN_ + k + kh + 1];
            v2f bA; bA.x = W1[(k+kh)*H_ + n0];            bA.y = W1[(k+kh+1)*H_ + n0];
            v2f bB; bB.x = W1[(k+kh)*H_ + n1];            bB.y = W1[(k+kh+1)*H_ + n1];
            acc00 = wmma_f32(a0, bA, acc00);
            acc01 = wmma_f32(a0, bB, acc01);
            acc10 = wmma_f32(a1, bA, acc10);
            acc11 = wmma_f32(a1, bB, acc11);
        }
        float bn0 = bias1[n0], bn1 = bias1[n1];
        #pragma unroll
        for (int r = 0; r < 8; ++r) {
            int M = r + mh;
            float v00 = acc00[r] + bn0, v01 = acc01[r] + bn1;
            float v10 = acc10[r] + bn0, v11 = acc11[r] + bn1;
            h1s[0*R_*H_ + M*H_ + n0] = v00 > 0.f ? v00 : 0.f;
            h1s[0*R_*H_ + M*H_ + n1] = v01 > 0.f ? v01 : 0.f;
            h1s[1*R_*H_ + M*H_ + n0] = v10 > 0.f ? v10 : 0.f;
            h1s[1*R_*H_ + M*H_ + n1] = v11 > 0.f ? v11 : 0.f;
        }
    }
    __syncthreads();

    // ---- Step 6: LayerNorm over H_ (h1s) ----
    if (tid < G_*R_) {
        int g = tid >> 4, r = tid & 15;
        const float* row = h1s + g*R_*H_ + r*H_;
        float s = 0.f, ss = 0.f;
        for (int c = 0; c < H_; ++c) { float v = row[c]; s += v; ss += v*v; }
        float mu = s * (1.0f/H_);
        float var = ss * (1.0f/H_) - mu*mu;
        mu_s[tid] = mu; rs_s[tid] = rsqrtf(var + 1e-5f);
    }
    __syncthreads();
    #pragma unroll
    for (int j = 0; j < 32; ++j) {           // 256*32 == 8192 == 2*16*256
        int idx = tid*32 + j;
        int gr = idx >> 8, c = idx & 255;    // gr = g*16 + r
        h1s[idx] = (h1s[idx] - mu_s[gr]) * rs_s[gr] * ln1_g[c] + ln1_b[c];
    }
    __syncthreads();

    // ---- Step 7: GEMM2 (2x 16x256)@(256x256) fp32 WMMA, fused bias+ReLU ----
    {
        v8f acc00 = {}, acc01 = {}, acc10 = {}, acc11 = {};
        for (int k = 0; k < H_; k += 4) {
            v2f a0; a0.x = h1s[0*R_*H_ + m*H_ + k + kh]; a0.y = h1s[0*R_*H_ + m*H_ + k + kh + 1];
            v2f a1; a1.x = h1s[1*R_*H_ + m*H_ + k + kh]; a1.y = h1s[1*R_*H_ + m*H_ + k + kh + 1];
            v2f bA; bA.x = W2[(k+kh)*H_ + n0];           bA.y = W2[(k+kh+1)*H_ + n0];
            v2f bB; bB.x = W2[(k+kh)*H_ + n1];           bB.y = W2[(k+kh+1)*H_ + n1];
            acc00 = wmma_f32(a0, bA, acc00);
            acc01 = wmma_f32(a0, bB, acc01);
            acc10 = wmma_f32(a1, bA, acc10);
            acc11 = wmma_f32(a1, bB, acc11);
        }
        float bn0 = bias2[n0], bn1 = bias2[n1];
        #pragma unroll
        for (int r = 0; r < 8; ++r) {
            int M = r + mh;
            float v00 = acc00[r] + bn0, v01 = acc01[r] + bn1;
            float v10 = acc10[r] + bn0, v11 = acc11[r] + bn1;
            h2s[0*R_*H_ + M*H_ + n0] = v00 > 0.f ? v00 : 0.f;
            h2s[0*R_*H_ + M*H_ + n1] = v01 > 0.f ? v01 : 0.f;
            h2s[1*R_*H_ + M*H_ + n0] = v10 > 0.f ? v10 : 0.f;
            h2s[1*R_*H_ + M*H_ + n1] = v11 > 0.f ? v11 : 0.f;
        }
    }
    __syncthreads();

    // ---- Step 8: LayerNorm over H_ (h2s) ----
    if (tid < G_*R_) {
        int g = tid >> 4, r = tid & 15;
        const float* row = h2s + g*R_*H_ + r*H_;
        float s = 0.f, ss = 0.f;
        for (int c = 0; c < H_; ++c) { float v = row[c]; s += v; ss += v*v; }
        float mu = s * (1.0f/H_);
        float var = ss * (1.0f/H_) - mu*mu;
        mu_s[tid] = mu; rs_s[tid] = rsqrtf(var + 1e-5f);
    }
    __syncthreads();
    #pragma unroll
    for (int j = 0; j < 32; ++j) {
        int idx = tid*32 + j;
        int gr = idx >> 8, c = idx & 255;
        h2s[idx] = (h2s[idx] - mu_s[gr]) * rs_s[gr] * ln2_g[c] + ln2_b[c];
    }
    __syncthreads();

    // ---- Step 9: value head: (16x256) @ (256x1) + bv, per env ----
    for (int g = 0; g < G_; ++g) {
        {
            int r = tid >> 4, s16 = tid & 15;
            float p = 0.f;
            #pragma unroll
            for (int j = 0; j < 16; ++j) {
                int c = s16 + j*16;
                p += h2s[g*R_*H_ + r*H_ + c] * Wv[c];
            }
            red[tid] = p;
        }
        __syncthreads();
        if (tid < R_) {
            float v = 0.f;
            #pragma unroll
            for (int j = 0; j < 16; ++j) v += red[tid*16 + j];
            values[(e0+g)*R_ + tid] = v + bv[0];
        }
        __syncthreads();
    }
}

extern "C" void kernel_launch(void* const* d_in, const int* in_sizes, int n_in,
                              void* d_out, int out_size, void* d_ws, size_t ws_size,
                              hipStream_t stream) {
    const float* cent_obs = (const float*)d_in[0];
    const float* rnn      = (const float*)d_in[1];
    // d_in[2] = masks (unused by the value path)
    const float* ln_in_g  = (const float*)d_in[3];
    const float* ln_in_b  = (const float*)d_in[4];
    const float* W1       = (const float*)d_in[5];
    const float* b1       = (const float*)d_in[6];
    const float* ln1_g    = (const float*)d_in[7];
    const float* ln1_b    = (const float*)d_in[8];
    const float* W2       = (const float*)d_in[9];
    const float* b2       = (const float*)d_in[10];
    const float* ln2_g    = (const float*)d_in[11];
    const float* ln2_b    = (const float*)d_in[12];
    const float* Wv       = (const float*)d_in[13];
    const float* bv       = (const float*)d_in[14];
    float* out = (float*)d_out;

    // values -> out[0 .. B); one workgroup (8 wave32) per 2 envs (M=32 rows)
    critic_fused<<<E_/G_, 256, 0, stream>>>(cent_obs, ln_in_g, ln_in_b,
                                            W1, b1, ln1_g, ln1_b,
                                            W2, b2, ln2_g, ln2_b,
                                            Wv, bv, out);

    // rnn_states passthrough -> out[B .. B + B*H)
    hipMemcpyAsync(out + B_, rnn, (size_t)B_ * H_ * sizeof(float),
                   hipMemcpyDeviceToDevice, stream);
}